// Loss_89730456748401
// MI455X (gfx1250) — compile-verified
//
#include <hip/hip_runtime.h>
#include <stdint.h>
#include <math.h>

#define TDIM    8192
#define THREADS 256
#define ELEMS   (TDIM / THREADS)   // 32 elements per thread

typedef float v2f __attribute__((ext_vector_type(2)));
typedef float v8f __attribute__((ext_vector_type(8)));

// order-preserving float -> uint key (larger float => larger key)
__device__ __forceinline__ uint32_t f2key(uint32_t bits) {
    return bits ^ (uint32_t)(((int32_t)bits >> 31) | 0x80000000u);
}
__device__ __forceinline__ float key2f(uint32_t k) {
    uint32_t bits = (k & 0x80000000u) ? (k ^ 0x80000000u) : ~k;
    return __uint_as_float(bits);
}

// One workgroup per row: async-copy row to LDS, 4-pass radix select for the
// k-th largest, masked sum above threshold (+tie correction), per-row BCE term.
__global__ __launch_bounds__(THREADS)
void topk_row_loss_kernel(const float* __restrict__ scores,
                          const float* __restrict__ label,
                          const int*   __restrict__ seqlen,
                          float*       __restrict__ row_loss) {
    __shared__ float    lds_data[TDIM];       // 32 KB raw row
    __shared__ unsigned hist[256];            // radix histogram
    __shared__ unsigned bc_prefix;
    __shared__ int      bc_rem;
    __shared__ float    warp_sums[THREADS / 32];

    const int row = blockIdx.x;
    const int tid = threadIdx.x;
    const int L   = seqlen[row];              // scalar load
    const int k   = L / 16 + 1;

    if (tid == 0) { bc_prefix = 0u; bc_rem = k; }

    // ---- CDNA5 async DMA: global -> LDS, 8 x b128 per thread ----
    uint64_t gbase = (uint64_t)(uintptr_t)scores + (uint64_t)row * (TDIM * 4ull);
    uint32_t lds_base = (uint32_t)(uintptr_t)(&lds_data[0]);
    #pragma unroll
    for (int j = 0; j < ELEMS / 4; ++j) {
        uint32_t byte_off = (uint32_t)(j * (THREADS * 16) + tid * 16);
        uint32_t lds_addr = lds_base + byte_off;
        asm volatile("global_load_async_to_lds_b128 %0, %1, %2"
                     :: "v"(lds_addr), "v"(byte_off), "s"(gbase) : "memory");
    }
    asm volatile("s_wait_asynccnt 0" ::: "memory");
    __syncthreads();   // every wave waited its own ASYNCcnt; barrier makes all LDS visible

    // ---- 4-pass MSB-first radix select (exact k-th largest key) ----
    #pragma unroll 1
    for (int p = 0; p < 4; ++p) {
        hist[tid] = 0u;
        __syncthreads();
        const unsigned shift  = 24u - 8u * (unsigned)p;
        const uint64_t prefix = (uint64_t)bc_prefix;
        #pragma unroll
        for (int j = 0; j < ELEMS; ++j) {
            int idx = tid + j * THREADS;
            if (idx < L) {
                uint32_t u = f2key(__float_as_uint(lds_data[idx]));
                if (((uint64_t)u >> (shift + 8u)) == prefix)
                    atomicAdd(&hist[(u >> shift) & 0xFFu], 1u);
            }
        }
        __syncthreads();
        if (tid == 0) {
            int rem = bc_rem;
            unsigned g = 0u;
            int d = 255;
            for (; d > 0; --d) {
                unsigned h = hist[d];
                if (g + h >= (unsigned)rem) break;
                g += h;
            }
            bc_prefix = (unsigned)((bc_prefix << 8) | (unsigned)d);
            bc_rem    = rem - (int)g;
        }
        __syncthreads();
    }

    const uint32_t tkey = bc_prefix;   // exact key of k-th largest element
    const int      ties = bc_rem;      // how many copies of tkey are in the top-k

    // ---- sum of elements strictly greater than threshold ----
    float s = 0.0f;
    #pragma unroll
    for (int j = 0; j < ELEMS; ++j) {
        int idx = tid + j * THREADS;
        if (idx < L) {
            uint32_t bits = __float_as_uint(lds_data[idx]);
            if (f2key(bits) > tkey) s += __uint_as_float(bits);
        }
    }
    #pragma unroll
    for (int m = 16; m >= 1; m >>= 1) s += __shfl_xor(s, m, 32);
    if ((tid & 31) == 0) warp_sums[tid >> 5] = s;
    __syncthreads();

    if (tid == 0) {
        float tot = (float)ties * key2f(tkey);
        #pragma unroll
        for (int w = 0; w < THREADS / 32; ++w) tot += warp_sums[w];
        float inst = tot / (float)k;
        float lab  = label[row];
        // stable log-sigmoid: logsig(x) = min(x,0) - log1p(exp(-|x|))
        float ls      = log1pf(expf(-fabsf(inst)));
        float log_p   = fminf(inst, 0.0f)  - ls;
        float log_1mp = fminf(-inst, 0.0f) - ls;
        row_loss[row] = lab * log_p + (1.0f - lab) * log_1mp;
    }
}

// Final mean over B row losses using the matrix unit:
// A = ones(16x4), B packs 64 values per step, C accumulates column sums.
// D[i][j] = sum_k B[k][j]  (rows identical), so total = sum over 16 columns.
__global__ __launch_bounds__(32)
void wmma_reduce_kernel(const float* __restrict__ row_loss,
                        float* __restrict__ out, int B) {
    const int lane = threadIdx.x;            // exactly one wave32, EXEC all ones
    v8f c = {};
    const v2f a = {1.0f, 1.0f};              // ones A-matrix (16x4 f32 = 2 VGPRs)
    const int iters = B / 64;
    #pragma unroll 1
    for (int i = 0; i < iters; ++i) {
        v2f b;
        b.x = row_loss[i * 64 + lane];
        b.y = row_loss[i * 64 + 32 + lane];
        c = __builtin_amdgcn_wmma_f32_16x16x4_f32(
                /*neg_a=*/false, a, /*neg_b=*/false, b,
                /*c_mod=*/(short)0, c, /*reuse_a=*/false, /*reuse_b=*/false);
    }
    // c[0]: lanes 0..15 hold D[0][j] (lanes 16..31 duplicate row 8 == row 0)
    float s = c[0];
    #pragma unroll
    for (int m = 8; m >= 1; m >>= 1) s += __shfl_xor(s, m, 32);
    if (lane == 0) out[0] = -s / (float)B;
}

extern "C" void kernel_launch(void* const* d_in, const int* in_sizes, int n_in,
                              void* d_out, int out_size, void* d_ws, size_t ws_size,
                              hipStream_t stream) {
    const float* scores = (const float*)d_in[0];
    const float* label  = (const float*)d_in[1];
    const int*   seqlen = (const int*)d_in[2];
    float* out      = (float*)d_out;
    float* row_loss = (float*)d_ws;           // B floats of scratch
    const int B = in_sizes[1];                // 2048 (multiple of 64)

    topk_row_loss_kernel<<<B, THREADS, 0, stream>>>(scores, label, seqlen, row_loss);
    wmma_reduce_kernel<<<1, 32, 0, stream>>>(row_loss, out, B);
}